// GATConv_21466246546036
// MI455X (gfx1250) — compile-verified
//
#include <hip/hip_runtime.h>
#include <hip/hip_bf16.h>

#define N_NODES   100000
#define N_EDGES   1600000
#define IN_F      256
#define OUT_F     128
#define NUM_HEADS 4
#define ATTR_F    64          // QUERY_DIM * NUM_HEADS
#define HEAD_D    (OUT_F / NUM_HEADS)   // 32

typedef __attribute__((ext_vector_type(2))) float v2f;
typedef __attribute__((ext_vector_type(8))) float v8f;

__device__ __forceinline__ void atomic_add_f32(float* p, float v) {
  __hip_atomic_fetch_add(p, v, __ATOMIC_RELAXED, __HIP_MEMORY_SCOPE_AGENT);
}

// ---------------------------------------------------------------------------
// Kernel 1: fused node GEMM via V_WMMA_F32_16X16X4_F32.
//   attr[n, 0:64]  = x @ W_w.T  + W_b
//   hbuf[n, 0:128] = x @ lin_w.T + lin_b
// One block = 16 nodes x 192 output cols. 12 waves, each wave owns one 16x16
// tile (wave id selects the N-tile; waves 0-3 -> attr cols, 4-11 -> h cols).
// A fragment: lanes 0-15 hold K=0,1 ; lanes 16-31 hold K=2,3 (ISA 7.12.2).
// B fragment mirrors A with N in lane[3:0]. C/D: VGPR j = M=j / M=8+j.
// Destination (buffer, stride) selected ONCE before the epilogue so the
// store loop is branch-free (previous version emitted per-element
// saveexec/predicated dual-store sequences).
// ---------------------------------------------------------------------------
__global__ __launch_bounds__(384)
void gat_node_gemm(const float* __restrict__ x,
                   const float* __restrict__ W_w, const float* __restrict__ W_b,
                   const float* __restrict__ lin_w, const float* __restrict__ lin_b,
                   float* __restrict__ attr, float* __restrict__ hbuf) {
  const int wave = threadIdx.x >> 5;            // 0..11 : N tile
  const int lane = threadIdx.x & 31;
  const int node_base = blockIdx.x << 4;        // 100000 % 16 == 0, no tail
  const int col  = (wave << 4) + (lane & 15);   // output column 0..191
  const int koff = (lane >> 4) << 1;            // 0 for lanes 0-15, 2 for 16-31

  const float* xrow = x + (size_t)(node_base + (lane & 15)) * IN_F;

  // Wave-uniform selection of weight row / bias / destination.
  const float* wrow;
  float bias;
  float* dst;          // points at [node_base + mbase, local col]
  size_t dstride;      // floats per row in destination buffer
  const int mbase = (lane >> 4) << 3;           // lanes 16-31 hold M=8..15
  if (col < ATTR_F) {
    wrow    = W_w + (size_t)col * IN_F;
    bias    = W_b[col];
    dst     = attr + (size_t)(node_base + mbase) * ATTR_F + col;
    dstride = ATTR_F;
  } else {
    const int c = col - ATTR_F;
    wrow    = lin_w + (size_t)c * IN_F;
    bias    = lin_b[c];
    dst     = hbuf + (size_t)(node_base + mbase) * OUT_F + c;
    dstride = OUT_F;
  }

  v8f acc = {};                                 // 16x16 f32 accumulator tile
#pragma unroll 8
  for (int k = 0; k < IN_F; k += 4) {
    v2f a = *(const v2f*)(xrow + k + koff);     // A[M=lane%16][K=k+koff, +1]
    v2f b = *(const v2f*)(wrow + k + koff);     // B[K=k+koff, +1][N=lane%16]
    acc = __builtin_amdgcn_wmma_f32_16x16x4_f32(
        /*neg_a=*/false, a, /*neg_b=*/false, b,
        /*c_mod=*/(short)0, acc, /*reuse_a=*/false, /*reuse_b=*/false);
  }

#pragma unroll
  for (int j = 0; j < 8; ++j) {
    dst[(size_t)j * dstride] = acc[j] + bias;   // branch-free epilogue
  }
}

// ---------------------------------------------------------------------------
// Kernel 2: per-node attention scores.
//   score_src[n,h] = attr[n,:] . a_w[h, 0:64]
//   score_dst[n,h] = attr[n,:] . a_w[h, 64:128] + a_b[h]
// ---------------------------------------------------------------------------
__global__ __launch_bounds__(256)
void gat_scores(const float* __restrict__ attr,
                const float* __restrict__ a_w, const float* __restrict__ a_b,
                float* __restrict__ score_src, float* __restrict__ score_dst) {
  const int t = blockIdx.x * blockDim.x + threadIdx.x;
  if (t >= N_NODES * 8) return;
  const int node  = t >> 3;
  const int head  = (t >> 1) & 3;
  const int isdst = t & 1;
  const float* ar = attr + (size_t)node * ATTR_F;
  const float* aw = a_w + head * (2 * ATTR_F) + isdst * ATTR_F;
  float sum = 0.f;
#pragma unroll 8
  for (int q = 0; q < ATTR_F; ++q) sum += aw[q] * ar[q];
  if (isdst) score_dst[node * NUM_HEADS + head] = sum + a_b[head];
  else       score_src[node * NUM_HEADS + head] = sum;
}

// ---------------------------------------------------------------------------
// Kernel 3: zero the accumulator (d_out). float4 stores.
// ---------------------------------------------------------------------------
__global__ __launch_bounds__(256)
void gat_zero(float4* __restrict__ out, int n4) {
  const int t = blockIdx.x * blockDim.x + threadIdx.x;
  if (t < n4) out[t] = make_float4(0.f, 0.f, 0.f, 0.f);
}

// ---------------------------------------------------------------------------
// Kernel 4: edge scatter. One wave per edge; lane handles float4 (128 cols).
//   alpha[h] = score_src[s,h] + score_dst[r,h]
//   out[r, c] += h[s, c] * alpha[c / 32]     (atomic f32 add)
// ---------------------------------------------------------------------------
__global__ __launch_bounds__(256)
void gat_edge_scatter(const int* __restrict__ ei,
                      const float* __restrict__ hbuf,
                      const float* __restrict__ score_src,
                      const float* __restrict__ score_dst,
                      float* __restrict__ out) {
  const int edge = blockIdx.x * 8 + (threadIdx.x >> 5);
  if (edge >= N_EDGES) return;                  // wave-uniform
  const int lane = threadIdx.x & 31;
  const int s = ei[edge];                       // edge_index[0, e]
  const int r = ei[N_EDGES + edge];             // edge_index[1, e]
  const int c0   = lane << 2;                   // 4 floats per lane
  const int head = c0 >> 5;                     // HEAD_D == 32
  const float alpha = score_src[s * NUM_HEADS + head]
                    + score_dst[r * NUM_HEADS + head];
  const float4 hv = *(const float4*)(hbuf + (size_t)s * OUT_F + c0);
  float* dst = out + (size_t)r * OUT_F + c0;
  atomic_add_f32(dst + 0, hv.x * alpha);
  atomic_add_f32(dst + 1, hv.y * alpha);
  atomic_add_f32(dst + 2, hv.z * alpha);
  atomic_add_f32(dst + 3, hv.w * alpha);
}

// ---------------------------------------------------------------------------
// Kernel 5: SiLU in place on d_out.
// ---------------------------------------------------------------------------
__global__ __launch_bounds__(256)
void gat_silu(float* __restrict__ out, int n) {
  const int t = blockIdx.x * blockDim.x + threadIdx.x;
  if (t < n) {
    const float v = out[t];
    out[t] = v / (1.f + __expf(-v));
  }
}

extern "C" void kernel_launch(void* const* d_in, const int* in_sizes, int n_in,
                              void* d_out, int out_size, void* d_ws, size_t ws_size,
                              hipStream_t stream) {
  const float* x     = (const float*)d_in[0];
  const int*   ei    = (const int*)  d_in[1];   // (2, E), int32 per harness
  const float* W_w   = (const float*)d_in[2];
  const float* W_b   = (const float*)d_in[3];
  const float* a_w   = (const float*)d_in[4];
  const float* a_b   = (const float*)d_in[5];
  const float* lin_w = (const float*)d_in[6];
  const float* lin_b = (const float*)d_in[7];
  float* out = (float*)d_out;

  char* ws = (char*)d_ws;
  float* attr = (float*)ws;                                              // 25.6 MB
  float* hbuf = (float*)(ws + (size_t)N_NODES * ATTR_F * sizeof(float)); // 51.2 MB
  float* ssrc = (float*)(ws + (size_t)N_NODES * (ATTR_F + OUT_F) * sizeof(float));
  float* sdst = ssrc + (size_t)N_NODES * NUM_HEADS;

  // 1) fused WMMA GEMM: attr + h
  gat_node_gemm<<<N_NODES / 16, 384, 0, stream>>>(x, W_w, W_b, lin_w, lin_b,
                                                  attr, hbuf);
  // 2) per-node scores
  gat_scores<<<(N_NODES * 8 + 255) / 256, 256, 0, stream>>>(attr, a_w, a_b,
                                                            ssrc, sdst);
  // 3) zero accumulator
  gat_zero<<<(N_NODES * OUT_F / 4 + 255) / 256, 256, 0, stream>>>(
      (float4*)out, N_NODES * OUT_F / 4);
  // 4) edge gather-scale-scatter (dominant: ~2 GB traffic + 205M f32 atomics)
  gat_edge_scatter<<<(N_EDGES + 7) / 8, 256, 0, stream>>>(ei, hbuf, ssrc, sdst,
                                                          out);
  // 5) SiLU epilogue
  gat_silu<<<(N_NODES * OUT_F + 255) / 256, 256, 0, stream>>>(out,
                                                              N_NODES * OUT_F);
}